// RmseLoss_mcp_35270271434984
// MI455X (gfx1250) — compile-verified
//
#include <hip/hip_runtime.h>
#include <hip/hip_bf16.h>
#include <stddef.h>

// Problem geometry (fixed by the reference: (4096, 4096, 1))
#define TDIM 4096     // time steps
#define RDIM 4096     // columns
#define NCH  32       // time chunks
#define TC   128      // increments per chunk
#define CT   256      // columns per block (== blockDim.x)
#define GRP  16       // rows staged per LDS group (16 KB p + 16 KB s per buffer)
#define RECF 8        // floats per (column, chunk) record

#if defined(__has_include)
#  if __has_include(<hip/amd_detail/amd_gfx1250_TDM.h>)
#    define TDM_6ARG 1
#  endif
#endif

typedef unsigned int v4u __attribute__((ext_vector_type(4)));
typedef int          v8i __attribute__((ext_vector_type(8)));
typedef int          v4i __attribute__((ext_vector_type(4)));

// ---- CDNA5 async helpers ---------------------------------------------------
// LDS byte offset of a generic pointer: addrspacecast to AS3 then ptrtoint.
typedef __attribute__((address_space(3))) const void* lds_cptr_t;
__device__ __forceinline__ unsigned lds_off_of(const void* p) {
    return (unsigned)(unsigned long long)(lds_cptr_t)p;
}

// GLOBAL_LOAD_ASYNC_TO_LDS_B128: per-lane 16B global->LDS, tracked on ASYNCcnt.
__device__ __forceinline__ void async_ld_b128(unsigned lds_off, const void* gaddr) {
    asm volatile("global_load_async_to_lds_b128 %0, %1, off"
                 :: "v"(lds_off), "v"(gaddr)
                 : "memory");
}
#define WAIT_ASYNC(n) asm volatile("s_wait_asynccnt " #n ::: "memory")

// TENSOR_LOAD_TO_LDS: one TDM descriptor moves a 2D tile (rows x 256 f32,
// row stride 4096 elements) into LDS. Tracked on TENSORcnt.
// D# packing per CDNA5 ISA.
__device__ __forceinline__ void tdm_load_tile_f32(unsigned lds_addr,
                                                  const void* gaddr, int rows) {
    unsigned long long ga = (unsigned long long)gaddr;
    v4u g0;
    g0.x = 1u;                                             // count=1, user desc
    g0.y = lds_addr;                                       // lds_addr[31:0]
    g0.z = (unsigned)ga;                                   // global_addr[31:0]
    g0.w = (unsigned)((ga >> 32) & 0x01ffffffu) | (2u << 30); // addr[56:32]|type=2
    v8i g1 = (v8i){0, 0, 0, 0, 0, 0, 0, 0};
    g1[0] = 0x20000;                  // wg_mask=0, data_size=2 (4 bytes)
    g1[1] = (int)(0x1000u << 16);     // tensor_dim0 = 4096 (lo16 @ [31:16])
    g1[2] = (int)(0x1000u << 16);     // tensor_dim0 hi=0; tensor_dim1=4096 lo16
    g1[3] = (int)(256u << 16);        // tensor_dim1 hi=0; tile_dim0 = 256
    g1[4] = rows & 0xffff;            // tile_dim1 = rows; tile_dim2 = 0
    g1[5] = 4096;                     // tensor_dim0_stride lo32 = 4096
    v4i z4 = (v4i){0, 0, 0, 0};
#ifdef TDM_6ARG
    v8i z8 = (v8i){0, 0, 0, 0, 0, 0, 0, 0};
    __builtin_amdgcn_tensor_load_to_lds(g0, g1, z4, z4, z8, 0);
#else
    __builtin_amdgcn_tensor_load_to_lds(g0, g1, z4, z4, 0);
#endif
}

// ---- Phase 1: per-(column, chunk) segmented scan out of LDS ----------------
__global__ __launch_bounds__(CT) void rmse_seg_phase1(
        const float* __restrict__ p, const int* __restrict__ s,
        float* __restrict__ rec) {
    __shared__ __align__(16) float pbuf[2][GRP][CT];
    __shared__ __align__(16) int   sbuf[2][GRP][CT];

    const int tid   = threadIdx.x;
    const int chunk = blockIdx.x;          // 0..NCH-1
    const int col0  = blockIdx.y * CT;     // column block
    const int t0    = chunk * TC;          // first increment index
    int tEnd = t0 + TC;                    // exclusive end of increments
    if (tEnd > TDIM - 1) tEnd = TDIM - 1;
    const int nrows = tEnd - t0 + 1;       // rows t0..tEnd inclusive
    const int ngrp  = (nrows + GRP - 1) / GRP;

    // p tile: single TDM descriptor, issued by wave 0 only.
    auto issue_p = [&](int g, int buf) {
        int r0   = t0 + g * GRP;
        int rows = nrows - g * GRP; if (rows > GRP) rows = GRP;
        const float* gp = p + (size_t)r0 * RDIM + col0;
        tdm_load_tile_f32(lds_off_of(&pbuf[buf][0][0]), gp, rows);
    };
    // s tile: per-lane async b128 copies (4 per thread per group).
    auto issue_s = [&](int g, int buf) {
        const int r0 = t0 + g * GRP;
        #pragma unroll
        for (int j = 0; j < 4; ++j) {
            int f  = tid + j * CT;         // int4 index in [0, 1024)
            int e  = f * 4;                // element index in 16x256 tile
            int rr = e >> 8;
            int cc = e & 255;
            int gr = r0 + rr;
            if (gr > TDIM - 1) gr = TDIM - 1;   // clamp (pad rows, never read)
            const int* gs = s + (size_t)gr * RDIM + col0 + cc;
            async_ld_b128(lds_off_of(&sbuf[buf][rr][cc]), gs);
        }
    };

    const bool w0 = (tid < 32);
    if (w0) issue_p(0, 0);
    issue_s(0, 0);

    // scan state for this thread's column
    float run_sum = 0.f, run_cnt = 0.f;        // current open run
    float head_sum = 0.f, head_cnt = 0.f;      // run before first boundary
    float imean = 0.f, ivalid = 0.f;           // closed interior segments
    int   seen_bound = 0;
    float prev_p = 0.f; int prev_s = 0; int first = 1;

    for (int g = 0; g < ngrp; ++g) {
        const int buf = g & 1;
        const bool more = (g + 1 < ngrp);
        if (more) {
            if (w0) issue_p(g + 1, buf ^ 1);
            issue_s(g + 1, buf ^ 1);
        }
        if (w0) {                              // p tile ready?
            if (more) __builtin_amdgcn_s_wait_tensorcnt((short)1);
            else      __builtin_amdgcn_s_wait_tensorcnt((short)0);
        }
        if (more) { WAIT_ASYNC(4); }           // s tile ready (in-order retire)
        else      { WAIT_ASYNC(0); }
        __syncthreads();                       // publish LDS tile to all waves

        int rcnt = nrows - g * GRP;
        if (rcnt > GRP) rcnt = GRP;
        for (int k = 0; k < rcnt; ++k) {
            float cp = pbuf[buf][k][tid];
            int   cs = sbuf[buf][k][tid];
            if (!first) {
                if (cs == prev_s) {
                    float d = cp - prev_p;
                    run_sum += d * d; run_cnt += 1.f;
                } else {
                    if (!seen_bound) { head_sum = run_sum; head_cnt = run_cnt; seen_bound = 1; }
                    else if (run_cnt > 0.f) { imean += run_sum / run_cnt; ivalid += 1.f; }
                    run_sum = 0.f; run_cnt = 0.f;
                }
            }
            first = 0; prev_p = cp; prev_s = cs;
        }
        __syncthreads();                       // reads done before buffer reuse
    }

    float* r = rec + ((size_t)chunk * RDIM + (col0 + tid)) * RECF;
    r[0] = head_sum; r[1] = head_cnt;
    r[2] = run_sum;  r[3] = run_cnt;           // tail (open) run
    r[4] = imean;    r[5] = ivalid;
    r[6] = seen_bound ? 1.f : 0.f; r[7] = 0.f;
}

// ---- Phase 2: merge chunk records per column, global atomics ---------------
__global__ __launch_bounds__(256) void rmse_seg_phase2(
        const float* __restrict__ rec, float* __restrict__ acc) {
    const int col = blockIdx.x * blockDim.x + threadIdx.x;   // 0..RDIM-1
    float cs = 0.f, cc = 0.f, msum = 0.f, mval = 0.f;
    for (int ch = 0; ch < NCH; ++ch) {
        const float* r = rec + ((size_t)ch * RDIM + col) * RECF;
        float hs = r[0], hc = r[1], ts = r[2], tc = r[3];
        float im = r[4], iv = r[5], hb = r[6];
        if (hb > 0.5f) {
            float tot = cs + hs, c = cc + hc;  // carry + head closes a segment
            if (c > 0.f) { msum += tot / c; mval += 1.f; }
            msum += im; mval += iv;
            cs = ts; cc = tc;
        } else {
            cs += ts; cc += tc;                // whole chunk extends the carry
        }
    }
    if (cc > 0.f) { msum += cs / cc; mval += 1.f; }  // final open segment
    atomicAdd(&acc[0], msum);
    atomicAdd(&acc[1], mval);
}

__global__ void rmse_seg_zero(float* acc) { acc[0] = 0.f; acc[1] = 0.f; }

__global__ void rmse_seg_final(const float* __restrict__ acc, float* __restrict__ out) {
    out[0] = (acc[1] > 0.f) ? (acc[0] / acc[1]) : 0.f;
}

// ---- Launch ----------------------------------------------------------------
extern "C" void kernel_launch(void* const* d_in, const int* in_sizes, int n_in,
                              void* d_out, int out_size, void* d_ws, size_t ws_size,
                              hipStream_t stream) {
    (void)in_sizes; (void)n_in; (void)out_size; (void)ws_size;
    const float* p = (const float*)d_in[0];   // 'output' f32 (4096,4096,1)
    const int*   s = (const int*)d_in[1];     // 'results' i32 (4096,4096,1)
    float* acc = (float*)d_ws;                // 2-float accumulator
    float* rec = (float*)d_ws + 64;           // records @ +256 B (4 MB)

    rmse_seg_zero<<<1, 1, 0, stream>>>(acc);
    dim3 grid1(NCH, RDIM / CT);               // 32 x 16 blocks, 8 wave32s each
    rmse_seg_phase1<<<grid1, CT, 0, stream>>>(p, s, rec);
    rmse_seg_phase2<<<RDIM / 256, 256, 0, stream>>>(rec, acc);
    rmse_seg_final<<<1, 1, 0, stream>>>(acc, (float*)d_out);
}